// VecEdgesRead_58523224375828
// MI455X (gfx1250) — compile-verified
//
#include <hip/hip_runtime.h>

typedef __attribute__((ext_vector_type(2))) float v2f;
typedef __attribute__((ext_vector_type(8))) float v8f;

#define INV_SQRT_2F 0.70710678118654752440f

// out[b,e,o,v] = (sum_i Wsrc[o,i]*x[b,src[e],i,v] + Wdst[o,i]*x[b,dst[e],i,v]) * INV_SQRT_2
//
// GEMM view: M=64 (o), K=128 (concat src/dst channel dims), N = 16 edges x 3 v-tiles.
// One wave handles 16 edges: 4 M-tiles x 3 v accumulators, K in steps of 4 via
// V_WMMA_F32_16X16X4_F32 (exact fp32, 1024 MACs/instr vs 32 for VALU FMA).
// The 1/sqrt(2) scale is folded into the LDS-staged weights (linearity), so the
// store epilogue is pure stores with no VALU scaling.
__global__ __launch_bounds__(256, 1)
void vec_edges_wmma(const float* __restrict__ x,
                    const float* __restrict__ Wsrc,
                    const float* __restrict__ Wdst,
                    const int* __restrict__ src,
                    const int* __restrict__ dst,
                    float* __restrict__ out,
                    int nodes, int edges)
{
    __shared__ float ldsW[64 * 128]; // [o][i'] concatenated pre-scaled weights, 32 KB

    // Cooperative load of INV_SQRT_2 * [W_src | W_dst] into LDS.
    for (int f = threadIdx.x; f < 64 * 128; f += 256) {
        const int o  = f >> 7;
        const int ip = f & 127;
        const float w = (ip < 64) ? Wsrc[(o << 6) + ip] : Wdst[(o << 6) + (ip - 64)];
        ldsW[f] = w * INV_SQRT_2F;
    }
    __syncthreads();

    const int lane = threadIdx.x & 31;
    const int wave = threadIdx.x >> 5;
    const int nIdx = lane & 15;   // edge-in-group (B column) / A row-in-tile
    const int half = lane >> 4;   // lane half selects K pair (A/B) and M+8 (C/D)

    const int b  = blockIdx.y;
    const int eg = blockIdx.x * 8 + wave;
    int e = eg * 16 + nIdx;
    const bool live = (e < edges);
    if (!live) e = edges - 1;     // clamp loads; stores predicated later

    const int sN = src[e];
    const int dN = dst[e];
    const float* __restrict__ xs = x + ((size_t)b * nodes + sN) * (64 * 3);
    const float* __restrict__ xd = x + ((size_t)b * nodes + dN) * (64 * 3);

    v8f acc[4][3];
    {
        v8f z = {};
        #pragma unroll
        for (int mt = 0; mt < 4; ++mt)
            #pragma unroll
            for (int v = 0; v < 3; ++v)
                acc[mt][v] = z;
    }

    const int rowOff = nIdx * 128;   // A-row base in LDS (M-tile adds 16*128)

    // Two K passes: pass 0 = src half (W cols 0..63), pass 1 = dst half (64..127).
    #pragma unroll 1
    for (int pass = 0; pass < 2; ++pass) {
        const float* __restrict__ xb = pass ? xd : xs;
        const int wcol = pass * 64;
        #pragma unroll 4
        for (int ks = 0; ks < 16; ++ks) {
            const int kb = ks * 4;          // K base within this 64-wide half
            const int ka = kb + half * 2;   // this lane's first K element

            // A fragments (pre-scaled weights) from LDS: K, K+1 for o = mt*16 + nIdx.
            v2f a[4];
            #pragma unroll
            for (int mt = 0; mt < 4; ++mt) {
                const int base = (mt * 16) * 128 + rowOff + wcol + ka;
                a[mt].x = ldsW[base];
                a[mt].y = ldsW[base + 1];
            }

            // B fragments: x[node][ka..ka+1][v] for v=0..2 -> 6 consecutive floats.
            float bl[6];
            const float* __restrict__ p = xb + ka * 3;
            #pragma unroll
            for (int q = 0; q < 6; ++q) bl[q] = p[q];
            v2f bf[3];
            #pragma unroll
            for (int v = 0; v < 3; ++v) { bf[v].x = bl[v]; bf[v].y = bl[3 + v]; }

            #pragma unroll
            for (int mt = 0; mt < 4; ++mt)
                #pragma unroll
                for (int v = 0; v < 3; ++v)
                    acc[mt][v] = __builtin_amdgcn_wmma_f32_16x16x4_f32(
                        false, a[mt], false, bf[v],
                        (short)0, acc[mt][v], false, false);
        }
    }

    // C/D layout: VGPR r, lane -> o = mt*16 + 8*half + r, column = edge e.
    // 3 v-components per (o,e) are contiguous in memory -> wide stores.
    if (live) {
        float* __restrict__ ob = out + ((size_t)b * edges + e) * (64 * 3);
        #pragma unroll
        for (int mt = 0; mt < 4; ++mt) {
            #pragma unroll
            for (int r = 0; r < 8; ++r) {
                const int o = mt * 16 + 8 * half + r;
                ob[o * 3 + 0] = acc[mt][0][r];
                ob[o * 3 + 1] = acc[mt][1][r];
                ob[o * 3 + 2] = acc[mt][2][r];
            }
        }
    }
}

extern "C" void kernel_launch(void* const* d_in, const int* in_sizes, int n_in,
                              void* d_out, int out_size, void* d_ws, size_t ws_size,
                              hipStream_t stream)
{
    const float* x    = (const float*)d_in[0];
    const float* Wsrc = (const float*)d_in[1];
    const float* Wdst = (const float*)d_in[2];
    const int*   src  = (const int*)d_in[3];
    const int*   dst  = (const int*)d_in[4];
    float* out = (float*)d_out;

    const int edges = in_sizes[3];
    const int batch = (int)((long long)out_size / ((long long)edges * 64 * 3));
    const int nodes = (int)((long long)in_sizes[0] / ((long long)batch * 64 * 3));

    const int egroups = (edges + 15) / 16;          // 16 edges per wave
    dim3 grid((egroups + 7) / 8, batch);            // 8 waves per block
    dim3 block(256);
    vec_edges_wmma<<<grid, block, 0, stream>>>(x, Wsrc, Wdst, src, dst, out,
                                               nodes, edges);
}